// PathwayAttentionPooling_24180665876641
// MI455X (gfx1250) — compile-verified
//
#include <hip/hip_runtime.h>
#include <hip/hip_bf16.h>

// Problem constants (match reference setup_inputs()).
#define NPROT   20000
#define DDIM    256
#define TMEM    262144
#define GSEG    16384
#define NDRUGS  4096
#define SMAXCAP 2048   // max members per group buffered in LDS (avg is 16)

typedef __attribute__((ext_vector_type(2))) float v2f;
typedef __attribute__((ext_vector_type(8))) float v8f;

// ---------------------------------------------------------------------------
// K0: zero drug_sum [NDRUGS*DDIM] and drug_cnt [NDRUGS] (contiguous region).
// ---------------------------------------------------------------------------
__global__ void k_zero(float* __restrict__ p, int n) {
    int i = blockIdx.x * blockDim.x + threadIdx.x;
    if (i < n) p[i] = 0.0f;
}

// ---------------------------------------------------------------------------
// K1: segment lower bounds. member_seg is sorted; seg_start[g] = first index
// with member_seg[idx] >= g, for g in [0, GSEG] inclusive.
// ---------------------------------------------------------------------------
__global__ void k_bounds(const int* __restrict__ member_seg,
                         int* __restrict__ seg_start) {
    int g = blockIdx.x * blockDim.x + threadIdx.x;
    if (g > GSEG) return;
    int lo = 0, hi = TMEM;
    while (lo < hi) {
        int mid = (lo + hi) >> 1;
        if (member_seg[mid] < g) lo = mid + 1; else hi = mid;
    }
    seg_start[g] = lo;
}

// ---------------------------------------------------------------------------
// K2: per-group attention-softmax pooling, accumulated into per-drug sums.
// One 256-thread block (8 wave32) per group.
//   Stage A: wave w computes s_i = dot(ph_i, attn_w) for members i = w, w+8...
//            (lane-strided columns + shfl_xor butterfly reduction).
//   Stage B: wave 0 reduces max and sum(exp); weights written back to LDS.
//   Stage C: thread c (= column) accumulates pooled_c = sum_i w_i*ph[i][c]
//            (coalesced row reads from the L2-resident protein table), then
//            global_atomic_add_f32 into drug_sum[drug][c].
// ---------------------------------------------------------------------------
__global__ void k_group_pool(const float* __restrict__ protein_h,
                             const int*   __restrict__ member_idx,
                             const float* __restrict__ attn_w,
                             const int*   __restrict__ seg_start,
                             const int*   __restrict__ group_to_drug,
                             float* __restrict__ drug_sum,
                             float* __restrict__ drug_cnt) {
    __shared__ float s_w[SMAXCAP];
    __shared__ int   s_row[SMAXCAP];
    __shared__ float s_red[2];

    const int g   = blockIdx.x;
    const int beg = seg_start[g];
    const int end = seg_start[g + 1];
    int n = end - beg;
    if (n <= 0) return;                 // empty group: contributes nothing
    if (n > SMAXCAP) n = SMAXCAP;       // safety cap (unreachable for this data)

    const int tid  = threadIdx.x;
    const int wave = tid >> 5;
    const int lane = tid & 31;

    // --- Stage A: attention logits ---
    for (int i = wave; i < n; i += 8) {
        const int rowbase = member_idx[beg + i] * DDIM;
        if (lane == 0) s_row[i] = rowbase;
        float p = 0.0f;
#pragma unroll
        for (int c = lane; c < DDIM; c += 32)
            p += protein_h[rowbase + c] * attn_w[c];
#pragma unroll
        for (int off = 16; off > 0; off >>= 1)
            p += __shfl_xor(p, off, 32);
        if (lane == 0) s_w[i] = p;
    }
    __syncthreads();

    // --- Stage B: stable softmax normalization ---
    if (wave == 0) {
        float m = -3.402823466e38f;
        for (int i = lane; i < n; i += 32) m = fmaxf(m, s_w[i]);
#pragma unroll
        for (int off = 16; off > 0; off >>= 1)
            m = fmaxf(m, __shfl_xor(m, off, 32));
        float dsum = 0.0f;
        for (int i = lane; i < n; i += 32) dsum += __expf(s_w[i] - m);
#pragma unroll
        for (int off = 16; off > 0; off >>= 1)
            dsum += __shfl_xor(dsum, off, 32);
        if (lane == 0) { s_red[0] = m; s_red[1] = dsum; }
    }
    __syncthreads();
    const float m   = s_red[0];
    const float inv = 1.0f / s_red[1];
    for (int i = tid; i < n; i += 256)
        s_w[i] = __expf(s_w[i] - m) * inv;
    __syncthreads();

    // --- Stage C: weighted pool per column, atomic-accumulate per drug ---
    const int c = tid;                  // 0..255 == column
    float acc = 0.0f;
    for (int i = 0; i < n; ++i)
        acc += s_w[i] * protein_h[s_row[i] + c];

    const int drug = group_to_drug[g];
    atomicAdd(&drug_sum[drug * DDIM + c], acc);
    if (tid == 0) atomicAdd(&drug_cnt[drug], 1.0f);
}

// ---------------------------------------------------------------------------
// K3: drug_mean = drug_sum / max(drug_cnt, 1)
// ---------------------------------------------------------------------------
__global__ void k_mean(const float* __restrict__ drug_sum,
                       const float* __restrict__ drug_cnt,
                       float* __restrict__ drug_mean) {
    const int d = blockIdx.x;
    const int c = threadIdx.x;
    const float denom = fmaxf(drug_cnt[d], 1.0f);
    drug_mean[d * DDIM + c] = drug_sum[d * DDIM + c] / denom;
}

// ---------------------------------------------------------------------------
// K4: out = relu(drug_mean @ proj_W^T + proj_b) via V_WMMA_F32_16X16X4_F32.
// One wave per 16x16 output tile; 64 WMMA steps over K=256.
//   A (16x4 f32) layout: lane = (h<<4)|ml, VGPR j holds K = kb + 2h + j
//       -> v2f a = A[row0+ml][kb+2h .. kb+2h+1]
//   B (4x16 f32) layout (rows striped across lanes): element [K][n] with
//       n = ml, K = kb + 2h + j; B[k][n] = proj_W[col0+n][k] (row-contiguous)
//   C/D (16x16 f32): VGPR r, lane -> row = r + 8h, col = ml.
// Grid is exact (512 blocks x 8 waves = 4096 tiles): EXEC all-ones for WMMA.
// ---------------------------------------------------------------------------
__global__ void k_proj_wmma(const float* __restrict__ A,      // [NDRUGS, DDIM]
                            const float* __restrict__ W,      // [DDIM, DDIM]
                            const float* __restrict__ bias,   // [DDIM]
                            float* __restrict__ out) {        // [NDRUGS, DDIM]
    const int wave = threadIdx.x >> 5;
    const int lane = threadIdx.x & 31;
    const int tile = blockIdx.x * 8 + wave;     // 0..4095
    const int tm = tile >> 4;                   // 256 M-tiles
    const int tn = tile & 15;                   // 16  N-tiles
    const int row0 = tm << 4;
    const int col0 = tn << 4;
    const int h  = lane >> 4;                   // K-half select
    const int ml = lane & 15;                   // m for A, n for B/C/D

    const float* arow = A + (row0 + ml) * DDIM + 2 * h;
    const float* brow = W + (col0 + ml) * DDIM + 2 * h;

    v8f acc = {};
#pragma unroll
    for (int kb = 0; kb < DDIM; kb += 4) {
        v2f a = *(const v2f*)(arow + kb);
        v2f b = *(const v2f*)(brow + kb);
        acc = __builtin_amdgcn_wmma_f32_16x16x4_f32(
            /*neg_a=*/false, a, /*neg_b=*/false, b,
            /*c_mod=*/(short)0, acc, /*reuse_a=*/false, /*reuse_b=*/false);
    }

    const float bn = bias[col0 + ml];
#pragma unroll
    for (int r = 0; r < 8; ++r) {
        float v = acc[r] + bn;
        v = v > 0.0f ? v : 0.0f;
        out[(row0 + r + 8 * h) * DDIM + col0 + ml] = v;
    }
}

// ---------------------------------------------------------------------------
extern "C" void kernel_launch(void* const* d_in, const int* in_sizes, int n_in,
                              void* d_out, int out_size, void* d_ws, size_t ws_size,
                              hipStream_t stream) {
    const float* protein_h  = (const float*)d_in[0];
    const int*   member_idx = (const int*)  d_in[1];
    const int*   member_seg = (const int*)  d_in[2];
    const int*   g2d        = (const int*)  d_in[3];
    const float* attn_w     = (const float*)d_in[4];
    const float* proj_W     = (const float*)d_in[5];
    const float* proj_b     = (const float*)d_in[6];
    float* out = (float*)d_out;

    // Workspace layout (bytes):
    //   drug_sum : NDRUGS*DDIM f32        (4 MB)
    //   drug_cnt : NDRUGS f32             (16 KB)   -- contiguous w/ drug_sum
    //   drug_mean: NDRUGS*DDIM f32        (4 MB)
    //   seg_start: (GSEG+1) int           (64 KB)
    char* ws = (char*)d_ws;
    float* drug_sum  = (float*)ws;                         ws += (size_t)NDRUGS * DDIM * 4;
    float* drug_cnt  = (float*)ws;                         ws += (size_t)NDRUGS * 4;
    float* drug_mean = (float*)ws;                         ws += (size_t)NDRUGS * DDIM * 4;
    int*   seg_start = (int*)ws;

    // 0) zero accumulators (drug_sum + drug_cnt are contiguous)
    {
        const int n = NDRUGS * DDIM + NDRUGS;
        k_zero<<<(n + 255) / 256, 256, 0, stream>>>(drug_sum, n);
    }
    // 1) segment bounds via binary search over sorted member_seg
    k_bounds<<<(GSEG + 1 + 255) / 256, 256, 0, stream>>>(member_seg, seg_start);
    // 2) per-group softmax pooling -> per-drug atomic accumulation
    k_group_pool<<<GSEG, 256, 0, stream>>>(protein_h, member_idx, attn_w,
                                           seg_start, g2d, drug_sum, drug_cnt);
    // 3) per-drug mean over non-empty groups
    k_mean<<<NDRUGS, 256, 0, stream>>>(drug_sum, drug_cnt, drug_mean);
    // 4) final projection + bias + ReLU on the f32 WMMA path
    k_proj_wmma<<<(NDRUGS / 16) * (DDIM / 16) / 8, 256, 0, stream>>>(
        drug_mean, proj_W, proj_b, out);
}